// CustomEmbedding_71760313581646
// MI455X (gfx1250) — compile-verified
//
#include <hip/hip_runtime.h>
#include <stdint.h>

// Embedding gather: out[i, :] = weight[:, x[i]]  (weight is [768, 50257] f32, row-major)
// Pure gather, 0 FLOPs. HBM floor ~204MB @ 23.3TB/s ~= 9us; weight (154MB) is L2-resident (192MB L2).
// CDNA5 path: async LDS copies (ASYNCcnt) — gather scattered dwords into LDS, store b128 out of LDS.

#define VOCAB 50257
#define EMBED 768
#define WAVES_PER_BLOCK 4
#define BLOCKS 1024

__global__ __launch_bounds__(WAVES_PER_BLOCK * 32)
void embed_gather_async(const int* __restrict__ x,
                        const float* __restrict__ w,
                        float* __restrict__ out,
                        int ntok, int nwaves)
{
    // per-wave double buffer: 2 x 768 floats; 4 waves -> 24KB LDS (of 320KB/WGP)
    __shared__ __align__(16) float smem[WAVES_PER_BLOCK * 2 * EMBED];

    const int lane = threadIdx.x & 31;
    const int wib  = threadIdx.x >> 5;                       // wave-in-block 0..3
    const int wid  = (blockIdx.x << 2) | wib;                // global wave id

    // LDS byte offset of this wave's buffer (flat-pointer low 32 bits == LDS offset)
    const unsigned lds_wave = (unsigned)(size_t)(&smem[wib * 2 * EMBED]);

    int stage = 0;
    for (int i = wid; i < ntok; i += nwaves) {
        const int t = x[i];                                   // wave-uniform token id
        const unsigned sbuf = lds_wave + (unsigned)stage * (EMBED * 4u);

        // ---- gather phase: 24 async dword loads, e = k*32 + lane ----
        // global byte offset (vs SADDR=w): ((k*32+lane)*VOCAB + t) * 4   (max ~155MB, fits i32)
        const unsigned gbase = ((unsigned)lane * (unsigned)VOCAB + (unsigned)t) * 4u;
#pragma unroll
        for (int k = 0; k < EMBED / 32; ++k) {
            unsigned goff = gbase + (unsigned)k * (32u * (unsigned)VOCAB * 4u);
            unsigned loff = sbuf + (unsigned)lane * 4u + (unsigned)k * 128u;
            asm volatile("global_load_async_to_lds_b32 %0, %1, %2"
                         :: "v"(loff), "v"(goff), "s"(w) : "memory");
        }

        // all of this wave's async ops (incl. prior-iteration stores) must land
        asm volatile("s_wait_asynccnt 0x0" ::: "memory");

        // ---- writeback phase: 6 async b128 stores, fully coalesced 3KB row ----
        const unsigned orow = (unsigned)i * (EMBED * 4u) + (unsigned)lane * 16u;
#pragma unroll
        for (int m = 0; m < EMBED / 128; ++m) {
            unsigned ooff = orow + (unsigned)m * 512u;
            unsigned loff = sbuf + (unsigned)lane * 16u + (unsigned)m * 512u;
            asm volatile("global_store_async_from_lds_b128 %0, %1, %2"
                         :: "v"(ooff), "v"(loff), "s"(out) : "memory");
        }
        stage ^= 1;  // next iteration's gathers overlap these stores
    }
    asm volatile("s_wait_asynccnt 0x0" ::: "memory");
    // S_ENDPGM performs implicit wait-idle as well.
}

extern "C" void kernel_launch(void* const* d_in, const int* in_sizes, int n_in,
                              void* d_out, int out_size, void* d_ws, size_t ws_size,
                              hipStream_t stream)
{
    const int*   x = (const int*)d_in[0];     // [8, 2048] int32 token ids
    const float* w = (const float*)d_in[1];   // [768, 50257] f32
    float*     out = (float*)d_out;           // [8, 2048, 768] f32

    const int ntok   = in_sizes[0];           // 16384
    const int nwaves = BLOCKS * WAVES_PER_BLOCK;

    hipLaunchKernelGGL(embed_gather_async,
                       dim3(BLOCKS), dim3(WAVES_PER_BLOCK * 32), 0, stream,
                       x, w, out, ntok, nwaves);
}